// SparseSDFVQVAE_3504693314203
// MI455X (gfx1250) — compile-verified
//
#include <hip/hip_runtime.h>

typedef __attribute__((ext_vector_type(2))) float        v2f;
typedef __attribute__((ext_vector_type(8))) float        v8f;
typedef __attribute__((ext_vector_type(4))) unsigned int v4u;
typedef __attribute__((ext_vector_type(8))) int          v8i;
typedef __attribute__((ext_vector_type(4))) int          v4i;

#define NUM_E      8192
#define DIM        64
#define NVOX       32768
#define QOUT_ELEMS (NVOX * DIM)
#define IDX_BASE   (QOUT_ELEMS + 2)

#define ROWS_PER_BLOCK 128   // 8 waves x 16 rows
#define TILE_ROWS      32    // codebook rows staged per iteration (2 x 16-col WMMA tiles)
#define LDS_STRIDE     68    // floats per staged row (64 + 4 pad -> bank-spread, 16B aligned)
#define LDS_TILE       (TILE_ROWS * LDS_STRIDE)
#define NTILES         (NUM_E / TILE_ROWS)

#if defined(__has_builtin)
#if __has_builtin(__builtin_amdgcn_tensor_load_to_lds) && __has_builtin(__builtin_amdgcn_s_wait_tensorcnt)
#define HAVE_TDM 1
#endif
#endif
#ifndef HAVE_TDM
#define HAVE_TDM 0
#endif

// ---------------------------------------------------------------------------
// Kernel 0: e_sq[k] = sum_d codebook[k][d]^2
// ---------------------------------------------------------------------------
__global__ __launch_bounds__(256) void esq_kernel(const float* __restrict__ cb,
                                                  float* __restrict__ esq) {
    int k = blockIdx.x * 256 + threadIdx.x;
    const float4* row = (const float4*)(cb + (size_t)k * DIM);
    float s = 0.f;
#pragma unroll
    for (int i = 0; i < DIM / 4; ++i) {
        float4 v = row[i];
        s += v.x * v.x + v.y * v.y + v.z * v.z + v.w * v.w;
    }
    esq[k] = s;
}

// ---------------------------------------------------------------------------
// TDM: DMA one 32x64 f32 codebook tile into LDS with hardware row padding
// (pad 4 DWORDs after every 64 DWORDs -> LDS row stride 68 floats).
// D# built per CDNA5 ISA ch.8 (group0/group1; groups 2-3 zero for 2D tile).
// Toolchain uses the clang-23 6-arg builtin: (v4u, v8i, v4i, v4i, v8i, i32).
// ---------------------------------------------------------------------------
#if HAVE_TDM
typedef __attribute__((address_space(3))) float lds_float;

__device__ __forceinline__ void tdm_load_tile(const float* gsrc, unsigned lds_byte_off) {
    unsigned long long ga = (unsigned long long)(uintptr_t)gsrc;
    v4u g0;
    g0.x = 1u;                                                   // count=1, user mode
    g0.y = lds_byte_off;                                         // lds_addr
    g0.z = (unsigned)(ga & 0xffffffffu);                         // global_addr[31:0]
    g0.w = (unsigned)((ga >> 32) & 0x01ffffffu) | (2u << 30);    // global_addr[56:32] | type=2
    v8i g1;
    g1[0] = (2 << 16)        // data_size = 4 bytes
          | (1 << 20)        // pad_enable
          | (5 << 22)        // pad_interval: every 64 DWORDs
          | (3 << 25);       // pad_amount: 4 DWORDs
    g1[1] = (DIM << 16);     // tensor_dim0 = 64 (low 16 bits at [63:48])
    g1[2] = (NUM_E << 16);   // tensor_dim0 hi=0 ; tensor_dim1 low16 = 8192
    g1[3] = (DIM << 16);     // tensor_dim1 hi=0 ; tile_dim0 = 64
    g1[4] = TILE_ROWS;       // tile_dim1 = 32 ; tile_dim2 = 0 (2D)
    g1[5] = DIM;             // tensor_dim0_stride = 64 elements
    g1[6] = 0;
    g1[7] = 0;
    v4i zero4 = {0, 0, 0, 0};
    v8i zero8 = {0, 0, 0, 0, 0, 0, 0, 0};
    __builtin_amdgcn_tensor_load_to_lds(g0, g1, zero4, zero4, zero8, 0);
}
#endif

// ---------------------------------------------------------------------------
// Per-tile compute: two independent 16-deep f32 WMMA chains + running argmin.
// ---------------------------------------------------------------------------
__device__ __forceinline__ void tile_compute(const float* __restrict__ bbase,
                                             const v2f a[16],
                                             const float* __restrict__ esq,
                                             int n0, int l16, int kOff,
                                             float bestS[8], int bestI[8]) {
    v8f acc0 = {};
    v8f acc1 = {};
#pragma unroll
    for (int t = 0; t < 16; ++t) {
        float2 b0v = *(const float2*)(bbase + l16 * LDS_STRIDE + 4 * t + kOff);
        float2 b1v = *(const float2*)(bbase + (16 + l16) * LDS_STRIDE + 4 * t + kOff);
        v2f b0 = {b0v.x, b0v.y};
        v2f b1 = {b1v.x, b1v.y};
        acc0 = __builtin_amdgcn_wmma_f32_16x16x4_f32(false, a[t], false, b0,
                                                     (short)0, acc0, false, false);
        acc1 = __builtin_amdgcn_wmma_f32_16x16x4_f32(false, a[t], false, b1,
                                                     (short)0, acc1, false, false);
    }
    float e0 = esq[n0 + l16];
    float e1 = esq[n0 + 16 + l16];
    int   i0 = n0 + l16;
    int   i1 = n0 + 16 + l16;
#pragma unroll
    for (int j = 0; j < 8; ++j) {
        float s0 = e0 - 2.0f * acc0[j];
        if (s0 < bestS[j]) { bestS[j] = s0; bestI[j] = i0; }
        float s1 = e1 - 2.0f * acc1[j];
        if (s1 < bestS[j]) { bestS[j] = s1; bestI[j] = i1; }
    }
}

// ---------------------------------------------------------------------------
// Kernel 1: fused  score = e_sq - 2*(z @ cb^T)  +  argmin  via f32 WMMA,
// codebook tiles streamed through double-buffered LDS by the TDM.
// ---------------------------------------------------------------------------
__global__ __launch_bounds__(256) void argmin_kernel(const float* __restrict__ z,
                                                     const float* __restrict__ cb,
                                                     const float* __restrict__ esq,
                                                     float* __restrict__ out) {
    __shared__ float smem[2 * LDS_TILE];

    const int tid     = threadIdx.x;
    const int lane    = tid & 31;
    const int wave    = tid >> 5;
    const int rowBase = blockIdx.x * ROWS_PER_BLOCK + wave * 16;
    const int halfSel = lane >> 4;        // 0: lanes 0-15, 1: lanes 16-31
    const int l16     = lane & 15;
    const int kOff    = halfSel * 2;      // ISA 16x4 f32 A layout: K = {0,1} / {2,3}

#if HAVE_TDM
    const unsigned ldsBase = (unsigned)(unsigned long long)(lds_float*)&smem[0];
    if (wave == 0) tdm_load_tile(cb, ldsBase);   // prefetch tile 0 -> buffer 0
#endif

    // A fragments: z tile, one row per (lane&15), all 16 K-steps kept resident.
    v2f a[16];
    {
        const float* zr = z + (size_t)(rowBase + l16) * DIM;
#pragma unroll
        for (int t = 0; t < 16; ++t) {
            float2 v = *(const float2*)(zr + 4 * t + kOff);
            v2f av = {v.x, v.y};
            a[t] = av;
        }
    }

    float bestS[8];
    int   bestI[8];
#pragma unroll
    for (int j = 0; j < 8; ++j) { bestS[j] = 3.4e38f; bestI[j] = 0; }

#if HAVE_TDM
    if (wave == 0) __builtin_amdgcn_s_wait_tensorcnt(0);
    __syncthreads();

    for (int iter = 0; iter < NTILES; ++iter) {
        const int    n0  = iter * TILE_ROWS;
        const int    cur = iter & 1;
        const float* bbase = &smem[cur * LDS_TILE];

        // DMA next tile into the other buffer while we compute on this one.
        if (wave == 0 && iter + 1 < NTILES)
            tdm_load_tile(cb + (size_t)(n0 + TILE_ROWS) * DIM,
                          ldsBase + (unsigned)((cur ^ 1) * LDS_TILE * 4));

        tile_compute(bbase, a, esq, n0, l16, kOff, bestS, bestI);

        if (wave == 0) __builtin_amdgcn_s_wait_tensorcnt(0);
        __syncthreads();
    }
#else
    // Fallback: synchronous cooperative staging (round-1 path).
    for (int n0 = 0; n0 < NUM_E; n0 += TILE_ROWS) {
        __syncthreads();
#pragma unroll
        for (int i = 0; i < 2; ++i) {
            int f  = tid + i * 256;
            int r  = f >> 4;
            int c4 = f & 15;
            float4 v = *(const float4*)(cb + (size_t)(n0 + r) * DIM + c4 * 4);
            *(float4*)(&smem[r * LDS_STRIDE + c4 * 4]) = v;
        }
        __syncthreads();
        tile_compute(smem, a, esq, n0, l16, kOff, bestS, bestI);
    }
#endif

    // Reduce across the 16 lanes of each half (C/D layout: VGPR j <-> row j / j+8).
#pragma unroll
    for (int j = 0; j < 8; ++j) {
        float s  = bestS[j];
        int   bi = bestI[j];
#pragma unroll
        for (int off = 1; off < 16; off <<= 1) {
            float os = __shfl_xor(s, off, 32);
            int   oi = __shfl_xor(bi, off, 32);
            if (os < s || (os == s && oi < bi)) { s = os; bi = oi; }
        }
        if (l16 == 0) {
            int row = rowBase + j + 8 * halfSel;
            out[IDX_BASE + row] = (float)bi;
        }
    }
}

// ---------------------------------------------------------------------------
// Kernel 2: quantized[r] = codebook[idx[r]] ; per-block partial of sum (z-q)^2
// ---------------------------------------------------------------------------
__global__ __launch_bounds__(256) void gather_loss_kernel(const float* __restrict__ z,
                                                          const float* __restrict__ cb,
                                                          float* __restrict__ out,
                                                          float* __restrict__ partials) {
    __shared__ float red[8];
    int r   = blockIdx.x * 256 + threadIdx.x;
    int idx = (int)out[IDX_BASE + r];

    const float4* q4 = (const float4*)(cb + (size_t)idx * DIM);
    const float4* z4 = (const float4*)(z + (size_t)r * DIM);
    float4*       o4 = (float4*)(out + (size_t)r * DIM);

    float err = 0.f;
#pragma unroll
    for (int i = 0; i < DIM / 4; ++i) {
        float4 q  = q4[i];
        float4 zv = z4[i];
        o4[i] = q;   // straight-through: z + sg(q - z) == q numerically
        float dx = zv.x - q.x, dy = zv.y - q.y, dz = zv.z - q.z, dw = zv.w - q.w;
        err += dx * dx + dy * dy + dz * dz + dw * dw;
    }

    // deterministic wave + block reduction
#pragma unroll
    for (int off = 16; off >= 1; off >>= 1) err += __shfl_xor(err, off, 32);
    if ((threadIdx.x & 31) == 0) red[threadIdx.x >> 5] = err;
    __syncthreads();
    if (threadIdx.x == 0) {
        float s = 0.f;
#pragma unroll
        for (int i = 0; i < 8; ++i) s += red[i];
        partials[blockIdx.x] = s;
    }
}

// ---------------------------------------------------------------------------
// Kernel 3: deterministic tree-sum of 128 partials -> both loss scalars
// ---------------------------------------------------------------------------
__global__ __launch_bounds__(128) void finalize_kernel(const float* __restrict__ partials,
                                                       float* __restrict__ out) {
    __shared__ float sm[128];
    sm[threadIdx.x] = partials[threadIdx.x];
    __syncthreads();
    for (int s = 64; s >= 1; s >>= 1) {
        if (threadIdx.x < s) sm[threadIdx.x] += sm[threadIdx.x + s];
        __syncthreads();
    }
    if (threadIdx.x == 0) {
        float m = sm[0] * (1.0f / (float)(NVOX * DIM));
        out[QOUT_ELEMS]     = m;  // vq_loss         (== mean((q - z)^2))
        out[QOUT_ELEMS + 1] = m;  // commitment_loss (== mean((z - q)^2))
    }
}

// ---------------------------------------------------------------------------
extern "C" void kernel_launch(void* const* d_in, const int* in_sizes, int n_in,
                              void* d_out, int out_size, void* d_ws, size_t ws_size,
                              hipStream_t stream) {
    const float* z  = (const float*)d_in[0];   // [32768, 64] f32
    const float* cb = (const float*)d_in[1];   // [8192, 64]  f32
    float* out      = (float*)d_out;
    float* esq      = (float*)d_ws;            // [8192]
    float* partials = esq + NUM_E;             // [128]

    esq_kernel<<<NUM_E / 256, 256, 0, stream>>>(cb, esq);
    argmin_kernel<<<NVOX / ROWS_PER_BLOCK, 256, 0, stream>>>(z, cb, esq, out);
    gather_loss_kernel<<<NVOX / 256, 256, 0, stream>>>(z, cb, out, partials);
    finalize_kernel<<<1, 128, 0, stream>>>(partials, out);
}